// PAM_Module_64115271794853
// MI455X (gfx1250) — compile-verified
//
#include <hip/hip_runtime.h>

typedef __attribute__((ext_vector_type(2))) float v2f;
typedef __attribute__((ext_vector_type(4))) float v4f;
typedef __attribute__((ext_vector_type(8))) float v8f;

#define EPSF 1e-6f

__device__ __forceinline__ float softplus_f(float z) {
    return z > 20.0f ? z : log1pf(__expf(z));
}

__device__ __forceinline__ v8f wmma_f32(v2f a, v2f b, v8f c) {
    return __builtin_amdgcn_wmma_f32_16x16x4_f32(false, a, false, b, (short)0, c, false, false);
}

// CDNA5 async global->LDS DMA (per-lane 16B), tracked by ASYNCcnt.
__device__ __forceinline__ void async_b128(unsigned int lds_byte_off, const void* gptr) {
    asm volatile("global_load_async_to_lds_b128 %0, %1, off"
                 :: "v"(lds_byte_off), "v"(gptr)
                 : "memory");
}
__device__ __forceinline__ void wait_async0() {
    asm volatile("s_wait_asynccnt 0" ::: "memory");
}

// -----------------------------------------------------------------------------
// Kernel 1: fused Q/K projection GEMM:  P(128 x 4096) = [Wq;Wk](128x512) @ x_b(512x4096)
// Double-buffered LDS, async global->LDS loads overlapped with WMMA.
// grid = B * 64 (n-groups of 64), block = 256 (8 waves). Block tile M=128, N=64.
// -----------------------------------------------------------------------------
__global__ __launch_bounds__(256) void qk_proj_kernel(
    const float* __restrict__ x,
    const float* __restrict__ Wq, const float* __restrict__ bq,
    const float* __restrict__ Wk, const float* __restrict__ bk,
    float* __restrict__ QK)
{
    __shared__ float Ws[2][128 * 36];   // 128 rows x 32 k, stride 36 (144B rows, 16B aligned)
    __shared__ float Xs[2][32 * 68];    // 32 k x 64 n, stride 68 (272B rows)

    const int b     = blockIdx.x >> 6;
    const int nBase = (blockIdx.x & 63) * 64;
    const int tid   = threadIdx.x;
    const int wave  = tid >> 5;
    const int lane  = tid & 31;
    const int lane16 = lane & 15;
    const int half   = lane >> 4;
    const int koff   = half * 2;
    const int m0     = wave * 16;
    const size_t xB  = (size_t)b * 512 * 4096;

    const unsigned int wBase = (unsigned int)(uintptr_t)&Ws[0][0];
    const unsigned int xBase = (unsigned int)(uintptr_t)&Xs[0][0];

    auto issue = [&](int k0, int buf) {
#pragma unroll
        for (int j = 0; j < 4; ++j) {               // W tile: 128x32 = 1024 float4
            int f = tid + j * 256;
            int row = f >> 3, c4 = (f & 7) * 4;
            const float* src = (row < 64) ? (Wq + row * 512 + k0 + c4)
                                          : (Wk + (row - 64) * 512 + k0 + c4);
            async_b128(wBase + (unsigned int)(buf * (128 * 36) + row * 36 + c4) * 4u, src);
        }
#pragma unroll
        for (int j = 0; j < 2; ++j) {               // X tile: 32x64 = 512 float4
            int f = tid + j * 256;
            int row = f >> 4, c4 = (f & 15) * 4;
            async_b128(xBase + (unsigned int)(buf * (32 * 68) + row * 68 + c4) * 4u,
                       &x[xB + (size_t)(k0 + row) * 4096 + nBase + c4]);
        }
    };

    v8f acc[4] = {};

    issue(0, 0);
    wait_async0();
    __syncthreads();

    for (int c = 0; c < 16; ++c) {
        const int buf = c & 1;
        if (c + 1 < 16) issue((c + 1) * 32, buf ^ 1);   // DMA next tile while computing
        const float* Wb = &Ws[buf][0];
        const float* Xb = &Xs[buf][0];
#pragma unroll
        for (int kk = 0; kk < 32; kk += 4) {
            v2f a = *(const v2f*)&Wb[(m0 + lane16) * 36 + kk + koff];
#pragma unroll
            for (int t = 0; t < 4; ++t) {
                v2f bb;
                bb.x = Xb[(kk + koff    ) * 68 + t * 16 + lane16];
                bb.y = Xb[(kk + koff + 1) * 68 + t * 16 + lane16];
                acc[t] = wmma_f32(a, bb, acc[t]);
            }
        }
        wait_async0();
        __syncthreads();
    }

#pragma unroll
    for (int t = 0; t < 4; ++t)
#pragma unroll
        for (int r = 0; r < 8; ++r) {
            int m = m0 + r + half * 8;
            int n = nBase + t * 16 + lane16;
            float bias = (m < 64) ? bq[m] : bk[m - 64];
            QK[((size_t)b * 128 + m) * 4096 + n] = softplus_f(acc[t][r] + bias);
        }
}

// -----------------------------------------------------------------------------
// Kernel 2: Ksum[b][m] = sum_n K[b][m][n]   (raw, EPS added later)
// -----------------------------------------------------------------------------
__global__ __launch_bounds__(256) void ksum_kernel(
    const float* __restrict__ QK, float* __restrict__ Ksum)
{
    __shared__ float red[256];
    const int row = blockIdx.x;
    const int b = row >> 6, m = row & 63;
    const float* src = QK + ((size_t)b * 128 + 64 + m) * 4096;
    float s = 0.0f;
    for (int i = threadIdx.x; i < 4096; i += 256) s += src[i];
    red[threadIdx.x] = s;
    __syncthreads();
    for (int off = 128; off > 0; off >>= 1) {
        if (threadIdx.x < off) red[threadIdx.x] += red[threadIdx.x + off];
        __syncthreads();
    }
    if (threadIdx.x == 0) Ksum[row] = red[0];
}

// -----------------------------------------------------------------------------
// Kernel 3: KX[b](64 x 512) = K[b](64 x 4096) @ x[b]^T(4096 x 512)
// Double-buffered LDS, async global->LDS loads. Deterministic (no atomics).
// grid = B * 8 (c-groups of 64).
// -----------------------------------------------------------------------------
__global__ __launch_bounds__(256) void kx_kernel(
    const float* __restrict__ x, const float* __restrict__ QK,
    float* __restrict__ KX)
{
    __shared__ float Ks[2][64 * 36];    // m x n-chunk
    __shared__ float Xc[2][64 * 36];    // c x n-chunk

    const int b = blockIdx.x >> 3;
    const int cBase = (blockIdx.x & 7) * 64;
    const int tid = threadIdx.x;
    const int wave = tid >> 5, lane = tid & 31;
    const int lane16 = lane & 15, half = lane >> 4, koff = half * 2;
    const int m0  = (wave & 3) * 16;
    const int c0s = (wave >> 2) * 32;

    const unsigned int kBase = (unsigned int)(uintptr_t)&Ks[0][0];
    const unsigned int xcBase = (unsigned int)(uintptr_t)&Xc[0][0];

    auto issue = [&](int n0, int buf) {
#pragma unroll
        for (int j = 0; j < 2; ++j) {               // 64x32 each = 512 float4
            int f = tid + j * 256;
            int row = f >> 3, c4 = (f & 7) * 4;
            async_b128(kBase + (unsigned int)(buf * (64 * 36) + row * 36 + c4) * 4u,
                       &QK[((size_t)b * 128 + 64 + row) * 4096 + n0 + c4]);
            async_b128(xcBase + (unsigned int)(buf * (64 * 36) + row * 36 + c4) * 4u,
                       &x[((size_t)b * 512 + cBase + row) * 4096 + n0 + c4]);
        }
    };

    v8f acc[2] = {};

    issue(0, 0);
    wait_async0();
    __syncthreads();

    for (int c = 0; c < 128; ++c) {
        const int buf = c & 1;
        if (c + 1 < 128) issue((c + 1) * 32, buf ^ 1);
        const float* Kb = &Ks[buf][0];
        const float* Xb = &Xc[buf][0];
#pragma unroll
        for (int kk = 0; kk < 32; kk += 4) {
            v2f a = *(const v2f*)&Kb[(m0 + lane16) * 36 + kk + koff];
#pragma unroll
            for (int u = 0; u < 2; ++u) {
                v2f bb = *(const v2f*)&Xb[(c0s + u * 16 + lane16) * 36 + kk + koff];
                acc[u] = wmma_f32(a, bb, acc[u]);
            }
        }
        wait_async0();
        __syncthreads();
    }

#pragma unroll
    for (int u = 0; u < 2; ++u)
#pragma unroll
        for (int r = 0; r < 8; ++r) {
            int m = m0 + r + half * 8;
            int c = cBase + c0s + u * 16 + lane16;
            KX[((size_t)b * 64 + m) * 512 + c] = acc[u][r];
        }
}

// -----------------------------------------------------------------------------
// Kernel 4: KV[b](64 x 512) = KX[b](64 x 512) @ Wv^T(512 x 512) + Ksum[b] (outer) bv
// grid = B * 8 (c-groups of 64)
// -----------------------------------------------------------------------------
__global__ __launch_bounds__(256) void kv_kernel(
    const float* __restrict__ KX, const float* __restrict__ Wv,
    const float* __restrict__ bv, const float* __restrict__ Ksum,
    float* __restrict__ KV)
{
    __shared__ float KXs[64 * 36];   // m x c'-chunk
    __shared__ float Wvs[64 * 36];   // c x c'-chunk

    const int b = blockIdx.x >> 3;
    const int cBase = (blockIdx.x & 7) * 64;
    const int tid = threadIdx.x;
    const int wave = tid >> 5, lane = tid & 31;
    const int lane16 = lane & 15, half = lane >> 4, koff = half * 2;
    const int m0  = (wave & 3) * 16;
    const int c0s = (wave >> 2) * 32;

    v8f acc[2] = {};

    for (int k0 = 0; k0 < 512; k0 += 32) {
#pragma unroll
        for (int j = 0; j < 2; ++j) {
            int f = tid + j * 256;
            int row = f >> 3, c4 = (f & 7) * 4;
            *(v4f*)&KXs[row * 36 + c4] =
                *(const v4f*)&KX[((size_t)b * 64 + row) * 512 + k0 + c4];
            *(v4f*)&Wvs[row * 36 + c4] =
                *(const v4f*)&Wv[(size_t)(cBase + row) * 512 + k0 + c4];
        }
        __syncthreads();
#pragma unroll
        for (int kk = 0; kk < 32; kk += 4) {
            v2f a = *(const v2f*)&KXs[(m0 + lane16) * 36 + kk + koff];
#pragma unroll
            for (int u = 0; u < 2; ++u) {
                v2f bb = *(const v2f*)&Wvs[(c0s + u * 16 + lane16) * 36 + kk + koff];
                acc[u] = wmma_f32(a, bb, acc[u]);
            }
        }
        __syncthreads();
    }

#pragma unroll
    for (int u = 0; u < 2; ++u)
#pragma unroll
        for (int r = 0; r < 8; ++r) {
            int m = m0 + r + half * 8;
            int c = cBase + c0s + u * 16 + lane16;
            KV[((size_t)b * 64 + m) * 512 + c] = acc[u][r] + Ksum[b * 64 + m] * bv[c];
        }
}

// -----------------------------------------------------------------------------
// Kernel 5: out[b][c][n] = x + gamma * norm[n] * sum_m KV[b][m][c] * Q[b][m][n]
//           norm[n] = 1 / sum_m Q[b][m][n] * (Ksum[b][m] + EPS)
// grid = B * 4 (c-groups of 128) * 64 (n-groups of 64); block = 256 (8 waves).
// -----------------------------------------------------------------------------
__global__ __launch_bounds__(256) void final_kernel(
    const float* __restrict__ x, const float* __restrict__ QK,
    const float* __restrict__ KV, const float* __restrict__ Ksum,
    const float* __restrict__ gamma, float* __restrict__ out)
{
    __shared__ float KVs[64 * 132];  // m x 128 c, stride 132 (528B rows, 16B aligned)
    __shared__ float Qs[64 * 68];    // m x 64 n
    __shared__ float KsE[64];
    __shared__ float norm_s[64];

    const int b     = blockIdx.x >> 8;
    const int rem   = blockIdx.x & 255;
    const int cBase = (rem >> 6) * 128;
    const int nBase = (rem & 63) * 64;
    const int tid = threadIdx.x;
    const int wave = tid >> 5, lane = tid & 31;
    const int lane16 = lane & 15, half = lane >> 4, koff = half * 2;
    const int c0 = wave * 16;
    const float g = gamma[0];

#pragma unroll
    for (int j = 0; j < 8; ++j) {                   // KV tile: 64x128 = 2048 float4
        int f = tid + j * 256;
        int row = f >> 5, c4 = (f & 31) * 4;
        *(v4f*)&KVs[row * 132 + c4] =
            *(const v4f*)&KV[((size_t)b * 64 + row) * 512 + cBase + c4];
    }
#pragma unroll
    for (int j = 0; j < 4; ++j) {                   // Q tile: 64x64 = 1024 float4
        int f = tid + j * 256;
        int row = f >> 4, c4 = (f & 15) * 4;
        *(v4f*)&Qs[row * 68 + c4] =
            *(const v4f*)&QK[((size_t)b * 128 + row) * 4096 + nBase + c4];
    }
    if (tid < 64) KsE[tid] = Ksum[b * 64 + tid] + EPSF;
    __syncthreads();

    if (tid < 64) {                                 // per-pixel normalizer
        float s = 0.0f;
#pragma unroll 8
        for (int m = 0; m < 64; ++m) s += Qs[m * 68 + tid] * KsE[m];
        norm_s[tid] = 1.0f / s;
    }
    __syncthreads();

    v8f acc[4] = {};
#pragma unroll
    for (int kk = 0; kk < 64; kk += 4) {
        v2f a;
        a.x = KVs[(kk + koff    ) * 132 + c0 + lane16];
        a.y = KVs[(kk + koff + 1) * 132 + c0 + lane16];
#pragma unroll
        for (int t = 0; t < 4; ++t) {
            v2f bb;
            bb.x = Qs[(kk + koff    ) * 68 + t * 16 + lane16];
            bb.y = Qs[(kk + koff + 1) * 68 + t * 16 + lane16];
            acc[t] = wmma_f32(a, bb, acc[t]);
        }
    }

#pragma unroll
    for (int t = 0; t < 4; ++t)
#pragma unroll
        for (int r = 0; r < 8; ++r) {
            int c  = cBase + c0 + r + half * 8;
            int nl = t * 16 + lane16;
            size_t idx = ((size_t)b * 512 + c) * 4096 + nBase + nl;
            out[idx] = x[idx] + g * norm_s[nl] * acc[t][r];
        }
}

// -----------------------------------------------------------------------------
extern "C" void kernel_launch(void* const* d_in, const int* in_sizes, int n_in,
                              void* d_out, int out_size, void* d_ws, size_t ws_size,
                              hipStream_t stream) {
    (void)in_sizes; (void)n_in; (void)out_size; (void)ws_size;
    const float* x     = (const float*)d_in[0];
    const float* Wq    = (const float*)d_in[1];
    const float* bq    = (const float*)d_in[2];
    const float* Wk    = (const float*)d_in[3];
    const float* bk    = (const float*)d_in[4];
    const float* Wv    = (const float*)d_in[5];
    const float* bv    = (const float*)d_in[6];
    const float* gamma = (const float*)d_in[7];
    float* out = (float*)d_out;

    // workspace layout (floats)
    float* ws   = (float*)d_ws;
    float* QK   = ws;                      // 16*128*4096 = 8388608
    float* Ksum = ws + 8388608;            // 16*64      = 1024
    float* KX   = ws + 8389632;            // 16*64*512  = 524288
    float* KV   = ws + 8913920;            // 16*64*512  = 524288

    qk_proj_kernel<<<dim3(16 * 64), dim3(256), 0, stream>>>(x, Wq, bq, Wk, bk, QK);
    ksum_kernel   <<<dim3(16 * 64), dim3(256), 0, stream>>>(QK, Ksum);
    kx_kernel     <<<dim3(16 * 8),  dim3(256), 0, stream>>>(x, QK, KX);
    kv_kernel     <<<dim3(16 * 8),  dim3(256), 0, stream>>>(KX, Wv, bv, Ksum, KV);
    final_kernel  <<<dim3(16 * 256),dim3(256), 0, stream>>>(x, QK, KV, Ksum, gamma, out);
}